// MeshSDFLoss_43430709297165
// MI455X (gfx1250) — compile-verified
//
#include <hip/hip_runtime.h>
#include <hip/hip_bf16.h>

typedef float v2f __attribute__((ext_vector_type(2)));
typedef float v8f __attribute__((ext_vector_type(8)));

// Fast reciprocal: v_rcp_f32 + one Newton-Raphson step (~1 ulp).
__device__ __forceinline__ float rcp_nr(float x) {
  float r = __builtin_amdgcn_rcpf(x);
  float e = __builtin_fmaf(-x, r, 1.0f);
  return __builtin_fmaf(r, e, r);
}

// ---------------------------------------------------------------------------
// Kernel 1: gather triangle data into a 64B-per-face record:
//   r0 = {ab.x, ab.y, ab.z, c1=ab.a}
//   r1 = {ac.x, ac.y, ac.z, c2=ac.a}
//   r2 = { a.x,  a.y,  a.z, 0}
//   r3 = {c3=ab.b, c4=ac.b, c5=ab.c, c6=ac.c}
// ---------------------------------------------------------------------------
__global__ void prep_faces_kernel(const float* __restrict__ verts,
                                  const int*   __restrict__ faces,
                                  float4*      __restrict__ frec,
                                  int F, int V) {
  int n = blockIdx.x * blockDim.x + threadIdx.x;
  if (n >= F) return;
  int i0 = faces[3 * n + 0];
  int i1 = faces[3 * n + 1];
  int i2 = faces[3 * n + 2];
  i0 = (i0 < 0) ? 0 : (i0 >= V ? V - 1 : i0);
  i1 = (i1 < 0) ? 0 : (i1 >= V ? V - 1 : i1);
  i2 = (i2 < 0) ? 0 : (i2 >= V ? V - 1 : i2);
  float ax = verts[3 * i0 + 0], ay = verts[3 * i0 + 1], az = verts[3 * i0 + 2];
  float bx = verts[3 * i1 + 0], by = verts[3 * i1 + 1], bz = verts[3 * i1 + 2];
  float cx = verts[3 * i2 + 0], cy = verts[3 * i2 + 1], cz = verts[3 * i2 + 2];
  float abx = bx - ax, aby = by - ay, abz = bz - az;
  float acx = cx - ax, acy = cy - ay, acz = cz - az;
  frec[4 * n + 0] = make_float4(abx, aby, abz, abx * ax + aby * ay + abz * az);
  frec[4 * n + 1] = make_float4(acx, acy, acz, acx * ax + acy * ay + acz * az);
  frec[4 * n + 2] = make_float4(ax, ay, az, 0.f);
  frec[4 * n + 3] = make_float4(abx * bx + aby * by + abz * bz,
                                acx * bx + acy * by + acz * bz,
                                abx * cx + aby * cy + abz * cz,
                                acx * cx + acy * cy + acz * cz);
}

// ---------------------------------------------------------------------------
// Kernel 2: brute-force closest-triangle search.
// 8 waves / block; wave owns a 16-point strip; loops over faces 16 at a time.
// D1 = points . AB^T and D2 = points . AC^T via V_WMMA_F32_16X16X4_F32.
// All per-pair logic is branchless (cndmask selects, no EXEC manipulation).
// ---------------------------------------------------------------------------
__global__ void __launch_bounds__(256)
sdf_kernel(const float*  __restrict__ points,
           const float4* __restrict__ frec,
           float* __restrict__ out_dist,
           int*   __restrict__ out_assoc,
           int P, int F) {
  const int lane = threadIdx.x & 31;
  const int wave = threadIdx.x >> 5;
  const int sp   = blockIdx.x * 128 + wave * 16;   // strip of 16 points
  const int half = lane >> 4;                      // 0: K=0,1  1: K=2,3
  const int l15  = lane & 15;

  // ---- A matrix (16 points x K4), K = {x,y,z,0} --------------------------
  int mrow = sp + l15;
  mrow = (mrow >= P) ? (P - 1) : mrow;
  const float* pm = points + 3 * mrow;
  v2f A;
  A.x = half ? pm[2] : pm[0];
  A.y = half ? 0.f   : pm[1];

  // ---- point coords for the elementwise phase (row = j + 8*half) --------
  float px[8], py[8], pz[8];
#pragma unroll
  for (int j = 0; j < 8; ++j) {
    int r = sp + j + 8 * half;
    r = (r >= P) ? (P - 1) : r;
    px[j] = points[3 * r + 0];
    py[j] = points[3 * r + 1];
    pz[j] = points[3 * r + 2];
  }

  const float INF = __builtin_inff();
  float best[8];
  int   bidx[8];
#pragma unroll
  for (int j = 0; j < 8; ++j) { best[j] = INF; bidx[j] = 0x7fffffff; }

  const v8f zero = {0.f, 0.f, 0.f, 0.f, 0.f, 0.f, 0.f, 0.f};

  for (int f0 = 0; f0 < F; f0 += 16) {
    const int  n     = f0 + l15;
    const bool valid = (n < F);
    const int  nc    = valid ? n : (F - 1);
    const int  nn    = valid ? n : 0x7fffffff;   // sentinel kills tie-break
    const float4 r0 = frec[4 * nc + 0];   // ab, c1
    const float4 r1 = frec[4 * nc + 1];   // ac, c2
    const float4 r2 = frec[4 * nc + 2];   // a
    const float4 r3 = frec[4 * nc + 3];   // c3,c4,c5,c6

    // speculative prefetch of next tile's record (dropped if OOB)
    __builtin_prefetch(&frec[4 * (nc + 16)], 0, 3);

    // ---- B matrices (K4 x 16 faces) ------------------------------------
    v2f Bab, Bac;
    Bab.x = half ? r0.z : r0.x;  Bab.y = half ? 0.f : r0.y;
    Bac.x = half ? r1.z : r1.x;  Bac.y = half ? 0.f : r1.y;

    // D1[m][n] = ab_n . p_m ; D2[m][n] = ac_n . p_m
    v8f D1 = __builtin_amdgcn_wmma_f32_16x16x4_f32(
        false, A, false, Bab, (short)0, zero, false, false);
    v8f D2 = __builtin_amdgcn_wmma_f32_16x16x4_f32(
        false, A, false, Bac, (short)0, zero, false, false);

#pragma unroll
    for (int j = 0; j < 8; ++j) {
      const float D1j = D1[j], D2j = D2[j];
      const float d1 = D1j - r0.w;
      const float d2 = D2j - r1.w;
      const float d3 = D1j - r3.x;
      const float d4 = D2j - r3.y;
      const float d5 = D1j - r3.z;
      const float d6 = D2j - r3.w;

      const float vc = d1 * d4 - d3 * d2;
      const float vb = d5 * d2 - d1 * d6;
      const float va = d3 * d6 - d5 * d4;

      float s_ab = d1 - d3;               s_ab = (s_ab == 0.f) ? 1.f : s_ab;
      const float v_ab = d1 * rcp_nr(s_ab);
      float s_ac = d2 - d6;               s_ac = (s_ac == 0.f) ? 1.f : s_ac;
      const float w_ac = d2 * rcp_nr(s_ac);
      const float num_bc = d4 - d3;
      const float e_bc   = d5 - d6;
      float s_bc = num_bc + e_bc;         s_bc = (s_bc == 0.f) ? 1.f : s_bc;
      const float w_bc = num_bc * rcp_nr(s_bc);
      float denom = va + vb + vc;         denom = (denom == 0.f) ? 1.f : denom;
      const float inv_den = rcp_nr(denom);
      const float v_in = vb * inv_den;
      const float w_in = vc * inv_den;

      const bool m_a  = (d1 <= 0.f) && (d2 <= 0.f);
      const bool m_b  = (d3 >= 0.f) && (d4 <= d3);
      const bool m_ab = (vc <= 0.f) && (d1 >= 0.f) && (d3 <= 0.f);
      const bool m_c  = (d6 >= 0.f) && (d5 <= d6);
      const bool m_ac = (vb <= 0.f) && (d2 >= 0.f) && (d6 <= 0.f);
      const bool m_bc = (va <= 0.f) && (num_bc >= 0.f) && (e_bc >= 0.f);

      const float v = m_a  ? 0.f
                    : m_b  ? 1.f
                    : m_ab ? v_ab
                    : m_c  ? 0.f
                    : m_ac ? 0.f
                    : m_bc ? (1.f - w_bc)
                    : v_in;
      const float w = m_a  ? 0.f
                    : m_b  ? 0.f
                    : m_ab ? 0.f
                    : m_c  ? 1.f
                    : m_ac ? w_ac
                    : m_bc ? w_bc
                    : w_in;

      // closest = a + v*ab + w*ac
      const float ccx = __builtin_fmaf(w, r1.x, __builtin_fmaf(v, r0.x, r2.x));
      const float ccy = __builtin_fmaf(w, r1.y, __builtin_fmaf(v, r0.y, r2.y));
      const float ccz = __builtin_fmaf(w, r1.z, __builtin_fmaf(v, r0.z, r2.z));
      const float dx = px[j] - ccx;
      const float dy = py[j] - ccy;
      const float dz = pz[j] - ccz;
      float sq = __builtin_fmaf(dz, dz,
                 __builtin_fmaf(dy, dy, dx * dx));
      sq = valid ? sq : INF;

      // branchless min+argmin update
      const bool better =
          (sq < best[j]) || ((sq == best[j]) && (nn < bidx[j]));
      best[j] = better ? sq : best[j];
      bidx[j] = better ? nn : bidx[j];
    }
  }

  // ---- min+argmin across the 16 lanes of each half-wave ------------------
#pragma unroll
  for (int j = 0; j < 8; ++j) {
#pragma unroll
    for (int off = 8; off >= 1; off >>= 1) {
      float od = __shfl_xor(best[j], off, 32);
      int   oi = __shfl_xor(bidx[j], off, 32);
      const bool better =
          (od < best[j]) || ((od == best[j]) && (oi < bidx[j]));
      best[j] = better ? od : best[j];
      bidx[j] = better ? oi : bidx[j];
    }
    if (l15 == 0) {
      int row = sp + j + 8 * half;
      if (row < P) {
        out_dist[row]  = best[j];
        out_assoc[row] = bidx[j];
      }
    }
  }
}

// ---------------------------------------------------------------------------
// Kernel 3: deterministic fixed-order loss reduction (no float atomics).
// ---------------------------------------------------------------------------
__global__ void loss_reduce_kernel(const float* __restrict__ dist,
                                   float* __restrict__ loss, int P) {
  __shared__ float sm[256];
  float s = 0.f;
  for (int i = threadIdx.x; i < P; i += 256) s += dist[i];
  sm[threadIdx.x] = s;
  __syncthreads();
  for (int st = 128; st > 0; st >>= 1) {
    if ((int)threadIdx.x < st) sm[threadIdx.x] += sm[threadIdx.x + st];
    __syncthreads();
  }
  if (threadIdx.x == 0) loss[0] = sm[0];
}

// ---------------------------------------------------------------------------
extern "C" void kernel_launch(void* const* d_in, const int* in_sizes, int n_in,
                              void* d_out, int out_size, void* d_ws,
                              size_t ws_size, hipStream_t stream) {
  const float* verts  = (const float*)d_in[0];
  const int*   faces  = (const int*)d_in[1];
  const float* points = (const float*)d_in[2];
  const int V = in_sizes[0] / 3;
  const int F = in_sizes[1] / 3;
  const int P = in_sizes[2] / 3;

  float*  out   = (float*)d_out;          // [0]=loss, [1..P]=dist, then assoc
  float*  dist  = out + 1;
  int*    assoc = (int*)(out + 1 + P);
  float4* frec  = (float4*)d_ws;          // F * 64 bytes

  prep_faces_kernel<<<(F + 255) / 256, 256, 0, stream>>>(verts, faces, frec,
                                                         F, V);
  sdf_kernel<<<(P + 127) / 128, 256, 0, stream>>>(points, frec, dist, assoc,
                                                  P, F);
  loss_reduce_kernel<<<1, 256, 0, stream>>>(dist, out, P);
}